// AttentionDecoder_49082886259298
// MI455X (gfx1250) — compile-verified
//
#include <hip/hip_runtime.h>

#define D 128
#define MAXN 20

typedef __attribute__((ext_vector_type(2))) float v2f;
typedef __attribute__((ext_vector_type(8))) float v8f;

// -------------------------------------------------------------------------
// Kernel 1: W = wq @ wk^T   (128x128, K=128).  One block = 8 waves, each
// wave computes a 16x128 strip via V_WMMA_F32_16X16X4_F32.
//   A[m][k] = wq[m*D + k]          (row-major)
//   B[k][n] = wk[n*D + k]          (wk^T -> contiguous v2f load per lane)
// -------------------------------------------------------------------------
__global__ __launch_bounds__(256) void wqk_wmma(const float* __restrict__ wq,
                                                const float* __restrict__ wk,
                                                float* __restrict__ Wm) {
  const int wave = threadIdx.x >> 5;
  const int lane = threadIdx.x & 31;
  const int half = lane >> 4;   // K split: 0 -> K{0,1}, 1 -> K{2,3}
  const int lrow = lane & 15;
  const int m0 = wave * 16;

  v8f acc[8];
  v8f zero = 0.0f;
#pragma unroll
  for (int t = 0; t < 8; ++t) acc[t] = zero;

  const float* aRow = wq + (m0 + lrow) * D;
  for (int k0 = 0; k0 < D; k0 += 4) {
    const v2f a = *(const v2f*)(aRow + k0 + 2 * half);
#pragma unroll
    for (int t = 0; t < 8; ++t) {
      const int n = t * 16 + lrow;
      const v2f b = *(const v2f*)(wk + n * D + k0 + 2 * half);  // B[k][n] = wk[n][k]
      acc[t] = __builtin_amdgcn_wmma_f32_16x16x4_f32(
          false, a, false, b, (short)0, acc[t], false, false);
    }
  }
#pragma unroll
  for (int t = 0; t < 8; ++t)
#pragma unroll
    for (int v = 0; v < 8; ++v)
      Wm[(m0 + v + 8 * half) * D + t * 16 + lrow] = acc[t][v];
}

// -------------------------------------------------------------------------
// Kernel 2: r = state_embed @ W   ([rows x 128] @ [128 x 128], rows=16384).
// One wave per 16 output rows, 8 accumulator tiles, 256 WMMAs per wave.
// W is 64KB -> L1/L2 resident, so per-tile B reloads are cheap.
// -------------------------------------------------------------------------
__global__ __launch_bounds__(256) void se_gemm(const float* __restrict__ se,
                                               const float* __restrict__ Wm,
                                               float* __restrict__ r,
                                               int rows) {
  const int wave = blockIdx.x * (blockDim.x >> 5) + (threadIdx.x >> 5);
  const int m0 = wave * 16;
  if (m0 >= rows) return;  // wave-uniform guard: EXEC stays all-ones
  const int lane = threadIdx.x & 31;
  const int half = lane >> 4;
  const int lrow = lane & 15;

  v8f acc[8];
  v8f zero = 0.0f;
#pragma unroll
  for (int t = 0; t < 8; ++t) acc[t] = zero;

  const float* aRow = se + (size_t)(m0 + lrow) * D;
  for (int k0 = 0; k0 < D; k0 += 4) {
    const v2f a = *(const v2f*)(aRow + k0 + 2 * half);
#pragma unroll
    for (int t = 0; t < 8; ++t) {
      const int n = t * 16 + lrow;
      v2f b;
      b.x = Wm[(k0 + 2 * half) * D + n];
      b.y = Wm[(k0 + 2 * half + 1) * D + n];
      acc[t] = __builtin_amdgcn_wmma_f32_16x16x4_f32(
          false, a, false, b, (short)0, acc[t], false, false);
    }
  }
#pragma unroll
  for (int t = 0; t < 8; ++t)
#pragma unroll
    for (int v = 0; v < 8; ++v)
      r[(size_t)(m0 + v + 8 * half) * D + t * 16 + lrow] = acc[t][v];
}

// -------------------------------------------------------------------------
// Kernel 3: out[i] = dot(r[rev[i]/20], action_embed[i])   for i in [0,total).
// One wave32 per node: lane loads float4 (coalesced 512B/wave of the 105MB
// action_embed stream), 5-step shfl_xor reduction, lane 0 stores.
// -------------------------------------------------------------------------
__global__ __launch_bounds__(256) void node_scores(const float* __restrict__ r,
                                                   const float* __restrict__ ae,
                                                   const int* __restrict__ rev,
                                                   float* __restrict__ out,
                                                   int total) {
  const int node = blockIdx.x * (blockDim.x >> 5) + (threadIdx.x >> 5);
  if (node >= total) return;
  const int lane = threadIdx.x & 31;
  const int b = rev[node] / MAXN;  // graph index of this node
  const float4 av = *(const float4*)(ae + (size_t)node * D + lane * 4);
  const float4 rv = *(const float4*)(r + (size_t)b * D + lane * 4);
  float s = av.x * rv.x + av.y * rv.y + av.z * rv.z + av.w * rv.w;
#pragma unroll
  for (int off = 16; off > 0; off >>= 1) s += __shfl_xor(s, off, 32);
  if (lane == 0) out[node] = s;
}

// -------------------------------------------------------------------------
// Inputs (setup_inputs order):
//   0 state_embed [16384*128] f32
//   1 action_embed [204800*128] f32
//   2 wq [128*128] f32
//   3 wk [128*128] f32
//   4 gather_idx (unused: output only reads valid slots)
//   5 valid_mask (unused: valid slots always have mask 1)
//   6 rev_idx [204800] i32
// Output: [204800] f32 (shape [total,1]).
// Workspace: W (64KB) + r (8MB) in d_ws.
// -------------------------------------------------------------------------
extern "C" void kernel_launch(void* const* d_in, const int* in_sizes, int n_in,
                              void* d_out, int out_size, void* d_ws, size_t ws_size,
                              hipStream_t stream) {
  const float* se  = (const float*)d_in[0];
  const float* ae  = (const float*)d_in[1];
  const float* wq  = (const float*)d_in[2];
  const float* wk  = (const float*)d_in[3];
  const int*   rev = (const int*)d_in[6];

  float* Wm = (float*)d_ws;
  float* r  = (float*)d_ws + D * D;

  const int rows  = in_sizes[0] / D;  // 16384 graphs
  const int total = in_sizes[6];      // 204800 nodes

  wqk_wmma<<<1, 256, 0, stream>>>(wq, wk, Wm);

  const int gemmWaves = (rows + 15) / 16;
  se_gemm<<<(gemmWaves + 7) / 8, 256, 0, stream>>>(se, Wm, r, rows);

  node_scores<<<(total + 7) / 8, 256, 0, stream>>>(r, ae, rev, (float*)d_out, total);
}